// OFFSET_PRED_MODULE_47949014893242
// MI455X (gfx1250) — compile-verified
//
#include <hip/hip_runtime.h>
#include <hip/hip_bf16.h>
#include <math.h>

// ---- problem constants (match reference) ----
#define BB 2
#define NN 8192
#define CC 128
#define KS 120      // K_SAMP
#define KN 30       // K_NN
#define DPAD 128    // padded dist columns (120 -> 128)

typedef __attribute__((ext_vector_type(2))) float v2f;
typedef __attribute__((ext_vector_type(8))) float v8f;

// =====================================================================
// K1: replicate np.random.RandomState(1234).shuffle(arange(8192))[:120]
//     (randomkit MT19937 seeding + masked-rejection random_interval)
// =====================================================================
#define MT_N 624
#define MT_M 397

__global__ void k_indices(int* __restrict__ idx_out) {
  __shared__ unsigned mt[MT_N];
  __shared__ int arr[NN];
  if (threadIdx.x == 0) {
    unsigned seed = 1234u;
    for (int pos = 0; pos < MT_N; ++pos) {
      mt[pos] = seed;
      seed = 1812433253u * (seed ^ (seed >> 30)) + (unsigned)pos + 1u;
    }
    int mpos = MT_N;
    for (int i = 0; i < NN; ++i) arr[i] = i;
    for (int i = NN - 1; i > 0; --i) {
      unsigned maxv = (unsigned)i;
      unsigned mask = maxv;
      mask |= mask >> 1; mask |= mask >> 2; mask |= mask >> 4;
      mask |= mask >> 8; mask |= mask >> 16;
      unsigned j;
      do {
        if (mpos == MT_N) {  // regenerate (single modulo loop == standard 3-loop)
          for (int k = 0; k < MT_N; ++k) {
            unsigned y = (mt[k] & 0x80000000u) | (mt[(k + 1) % MT_N] & 0x7fffffffu);
            unsigned v = mt[(k + MT_M) % MT_N] ^ (y >> 1);
            if (y & 1u) v ^= 0x9908b0dfu;
            mt[k] = v;
          }
          mpos = 0;
        }
        unsigned y = mt[mpos++];
        y ^= y >> 11; y ^= (y << 7) & 0x9d2c5680u;
        y ^= (y << 15) & 0xefc60000u; y ^= y >> 18;
        j = y & mask;
      } while (j > maxv);
      int t = arr[i]; arr[i] = arr[(int)j]; arr[(int)j] = t;
    }
    for (int i = 0; i < KS; ++i) idx_out[i] = arr[i];
  }
}

// =====================================================================
// K2: gather sampled keys; gn = normalized inst rows; copy kp/feat rows
//     one wave per (b, j)
// =====================================================================
__global__ void k_prep(const float* __restrict__ inst, const float* __restrict__ pts,
                       const float* __restrict__ feat, const int* __restrict__ idx,
                       float* __restrict__ gn, float* __restrict__ kp, float* __restrict__ fs) {
  int wid  = (int)((blockIdx.x * blockDim.x + threadIdx.x) >> 5);
  int lane = threadIdx.x & 31;
  int b = wid / KS, j = wid % KS;
  int key = idx[j];
  const float* src = inst + ((size_t)b * NN + key) * CC;
  float v[4]; float ss = 0.f;
  for (int q = 0; q < 4; ++q) { v[q] = src[lane + 32 * q]; ss += v[q] * v[q]; }
  for (int m = 16; m >= 1; m >>= 1) ss += __shfl_xor(ss, m, 32);
  float rn = rsqrtf(ss);
  float* gdst = gn + ((size_t)b * KS + j) * CC;
  for (int q = 0; q < 4; ++q) gdst[lane + 32 * q] = v[q] * rn;
  const float* fsrc = feat + ((size_t)b * NN + key) * CC;
  float* fdst = fs + ((size_t)b * KS + j) * CC;
  for (int q = 0; q < 4; ++q) fdst[lane + 32 * q] = fsrc[lane + 32 * q];
  if (lane < 3) kp[((size_t)b * KS + j) * 3 + lane] = pts[((size_t)b * NN + key) * 3 + lane];
}

// =====================================================================
// K3: rnorm[b,n] = 1/||inst[b,n]||   (one wave per point)
// =====================================================================
__global__ void k_rnorm(const float* __restrict__ inst, float* __restrict__ rnorm) {
  int wid  = (int)((blockIdx.x * blockDim.x + threadIdx.x) >> 5);
  int lane = threadIdx.x & 31;
  const float* src = inst + (size_t)wid * CC;
  float ss = 0.f;
  for (int q = 0; q < 4; ++q) { float x = src[lane + 32 * q]; ss += x * x; }
  for (int m = 16; m >= 1; m >>= 1) ss += __shfl_xor(ss, m, 32);
  if (lane == 0) rnorm[wid] = rsqrtf(ss);
}

// =====================================================================
// K4: dist = (inst @ gn^T) * rnorm - 1  via V_WMMA_F32_16X16X4_F32
//     grid (8 ntiles, N/16 mtiles, B), one wave per 16x16 tile
// =====================================================================
__global__ __launch_bounds__(32) void k_dist_wmma(const float* __restrict__ inst,
                                                  const float* __restrict__ gn,
                                                  const float* __restrict__ rnorm,
                                                  float* __restrict__ dist) {
  int nt = blockIdx.x, mt = blockIdx.y, b = blockIdx.z;
  int lane = threadIdx.x;
  int half = lane >> 4, l = lane & 15;
  int rowA = (mt << 4) + l;
  int colB = (nt << 4) + l;
  int colC = colB < KS ? colB : 0;
  float bmask = colB < KS ? 1.f : 0.f;
  const float* Ab = inst + ((size_t)b * NN + rowA) * CC;
  const float* Bb = gn + ((size_t)b * KS + colC) * CC;
  v8f acc = {};
  for (int k = 0; k < CC; k += 4) {
    int kk = k + (half << 1);
    v2f a, bv;
    a[0] = Ab[kk];          a[1] = Ab[kk + 1];
    bv[0] = Bb[kk] * bmask; bv[1] = Bb[kk + 1] * bmask;
    acc = __builtin_amdgcn_wmma_f32_16x16x4_f32(false, a, false, bv,
                                                (short)0, acc, false, false);
  }
  int col = (nt << 4) + l;
  for (int v = 0; v < 8; ++v) {
    int m = (mt << 4) + v + (half << 3);
    float rn = rnorm[(size_t)b * NN + m];
    dist[((size_t)b * NN + m) * DPAD + col] = acc[v] * rn - 1.0f;
  }
}

// =====================================================================
// K5: top-30 (sorted desc, stable) + kpam MLP + softmax -> attn
//     one wave per point
// =====================================================================
__global__ void k_topk_attn(const float* __restrict__ dist, const float* __restrict__ w1,
                            const float* __restrict__ w2, int* __restrict__ tidx,
                            float* __restrict__ attn) {
  __shared__ float td_s[4][KN];
  __shared__ float a_s[4][KN];
  int warp = threadIdx.x >> 5, lane = threadIdx.x & 31;
  int wid = blockIdx.x * 4 + warp;
  const float* drow = dist + (size_t)wid * DPAD;
  float d[4];
  for (int q = 0; q < 4; ++q) {
    int k = lane + 32 * q;
    d[q] = (k < KS) ? drow[k] : -3.0e38f;
  }
  for (int i = 0; i < KN; ++i) {
    float bvv = d[0]; int bk = lane;
    for (int q = 1; q < 4; ++q) {
      if (d[q] > bvv) { bvv = d[q]; bk = lane + 32 * q; }
    }
    for (int m = 16; m >= 1; m >>= 1) {
      float ov = __shfl_xor(bvv, m, 32);
      int   ok = __shfl_xor(bk, m, 32);
      if (ov > bvv || (ov == bvv && ok < bk)) { bvv = ov; bk = ok; }
    }
    if (lane == 0) {
      td_s[warp][i] = bvv;
      tidx[(size_t)wid * KN + i] = bk;
    }
    if ((bk & 31) == lane) d[bk >> 5] = -3.0e38f;
  }
  __syncthreads();
  float a1 = 0.f;
  if (lane < KN) {
    for (int j = 0; j < KN; ++j) a1 += w1[lane * KN + j] * td_s[warp][j];
    a1 = fmaxf(a1, 0.f);
    a_s[warp][lane] = a1;
  }
  __syncthreads();
  float a2 = -3.0e38f;
  if (lane < KN) {
    float s = 0.f;
    for (int o = 0; o < KN; ++o) s += w2[lane * KN + o] * a_s[warp][o];
    a2 = s;
  }
  float mx = a2;
  for (int m = 16; m >= 1; m >>= 1) mx = fmaxf(mx, __shfl_xor(mx, m, 32));
  float e = (lane < KN) ? expf(a2 - mx) : 0.f;
  float se = e;
  for (int m = 16; m >= 1; m >>= 1) se += __shfl_xor(se, m, 32);
  if (lane < KN) attn[(size_t)wid * KN + lane] = e / se;
}

// =====================================================================
// K6: fused  fa = attn*x ; y = fa @ conv_w^T  (WMMA f32 16x16x4)
//     emits per-channel max/min over k + per-(b,group) sum/sumsq.
//     Never materializes y (saves ~250 MB of HBM traffic).
//     one wave per point; fa tile staged in LDS (30x132, col 131 zero-pad)
// =====================================================================
__global__ __launch_bounds__(128) void k_conv_wmma(
    const float* __restrict__ pts, const float* __restrict__ convw,
    const int* __restrict__ tidx, const float* __restrict__ attn,
    const float* __restrict__ kp, const float* __restrict__ fs,
    float* __restrict__ ymax, float* __restrict__ ymin,
    float* __restrict__ gsum, float* __restrict__ gss) {
  __shared__ float fa[4][KN * 132];
  int warp = threadIdx.x >> 5, lane = threadIdx.x & 31;
  int wid = blockIdx.x * 4 + warp;
  int b = wid / NN, n = wid % NN;
  float px = pts[(size_t)wid * 3 + 0];
  float py = pts[(size_t)wid * 3 + 1];
  float pz = pts[(size_t)wid * 3 + 2];
  const int*   ti = tidx + (size_t)wid * KN;
  const float* at = attn + (size_t)wid * KN;
  for (int r = 0; r < KN; ++r) {
    int key = ti[r];
    float w = at[r];
    const float* frow = fs + ((size_t)b * KS + key) * CC;
    const float* krow = kp + ((size_t)b * KS + key) * 3;
    for (int c = lane; c < 132; c += 32) {
      float x;
      if (c < CC)        x = frow[c];
      else if (c == 128) x = krow[0] - px;
      else if (c == 129) x = krow[1] - py;
      else if (c == 130) x = krow[2] - pz;
      else               x = 0.f;
      fa[warp][r * 132 + c] = w * x;
    }
  }
  __syncthreads();

  int half = lane >> 4, l = lane & 15;
  float sum_g[2] = {0.f, 0.f}, ss_g[2] = {0.f, 0.f};
  for (int nt = 0; nt < 8; ++nt) {
    int o = (nt << 4) + l;      // output channel
    int g = nt >> 2;            // group (0: o<64, 1: o>=64)
    float lmax = -3.0e38f, lmin = 3.0e38f;
    const float* cw = convw + (size_t)o * 131;
    for (int mtl = 0; mtl < 2; ++mtl) {
      int row = (mtl << 4) + l;
      float rmask = (row < KN) ? 1.f : 0.f;
      const float* farow = &fa[warp][(row < KN ? row : 0) * 132];
      v8f acc = {};
      for (int ks = 0; ks < 33; ++ks) {
        int kk = (ks << 2) + (half << 1);
        v2f a, bv;
        a[0] = farow[kk] * rmask;
        a[1] = farow[kk + 1] * rmask;
        bv[0] = (kk < 131)     ? cw[kk]     : 0.f;
        bv[1] = (kk + 1 < 131) ? cw[kk + 1] : 0.f;
        acc = __builtin_amdgcn_wmma_f32_16x16x4_f32(false, a, false, bv,
                                                    (short)0, acc, false, false);
      }
      for (int v = 0; v < 8; ++v) {
        int kidx = (mtl << 4) + v + (half << 3);
        float val = acc[v];            // rows >= KN are exact zeros
        sum_g[g] += val;
        ss_g[g]  += val * val;
        if (kidx < KN) { lmax = fmaxf(lmax, val); lmin = fminf(lmin, val); }
      }
    }
    lmax = fmaxf(lmax, __shfl_xor(lmax, 16, 32));
    lmin = fminf(lmin, __shfl_xor(lmin, 16, 32));
    if (half == 0) {
      ymax[((size_t)b * CC + o) * NN + n] = lmax;
      ymin[((size_t)b * CC + o) * NN + n] = lmin;
    }
  }
  for (int m = 16; m >= 1; m >>= 1) {
    sum_g[0] += __shfl_xor(sum_g[0], m, 32);
    sum_g[1] += __shfl_xor(sum_g[1], m, 32);
    ss_g[0]  += __shfl_xor(ss_g[0], m, 32);
    ss_g[1]  += __shfl_xor(ss_g[1], m, 32);
  }
  if (lane == 0) {
    atomicAdd(&gsum[b * 2 + 0], sum_g[0]);
    atomicAdd(&gsum[b * 2 + 1], sum_g[1]);
    atomicAdd(&gss[b * 2 + 0], ss_g[0]);
    atomicAdd(&gss[b * 2 + 1], ss_g[1]);
  }
}

// =====================================================================
// K7: finalize GN statistics
// =====================================================================
__global__ void k_stats(const float* __restrict__ gsum, const float* __restrict__ gss,
                        float* __restrict__ gmu, float* __restrict__ gscale) {
  int t = threadIdx.x;
  if (t < 4) {
    float cnt = 64.f * KN * NN;   // (128/2 ch) * 30 * 8192
    float mu = gsum[t] / cnt;
    float var = gss[t] / cnt - mu * mu;
    gmu[t] = mu;
    gscale[t] = rsqrtf(var + 1e-5f);
  }
}

// =====================================================================
// K8: GN affine + leakyReLU + max-over-k (via max/min + sign of scale),
//     concat with feature, 3x256 MLP  -> offsets (B,3,N)
// =====================================================================
__global__ void k_final(const float* __restrict__ ymax, const float* __restrict__ ymin,
                        const float* __restrict__ feature, const float* __restrict__ gamma,
                        const float* __restrict__ beta, const float* __restrict__ mlpw,
                        const float* __restrict__ mlpb, const float* __restrict__ gmu,
                        const float* __restrict__ gscale, float* __restrict__ out) {
  int warp = threadIdx.x >> 5, lane = threadIdx.x & 31;
  int wid = blockIdx.x * 4 + warp;
  int b = wid / NN, n = wid % NN;
  float acc0 = 0.f, acc1 = 0.f, acc2 = 0.f;
  for (int q = 0; q < 4; ++q) {
    int c = lane + 32 * q;
    int g = c >> 6;
    float sc = gscale[b * 2 + g];
    float mu = gmu[b * 2 + g];
    float sg = gamma[c] * sc;
    float tt = beta[c] - mu * sg;
    float v = (sg >= 0.f) ? ymax[((size_t)b * CC + c) * NN + n]
                          : ymin[((size_t)b * CC + c) * NN + n];
    float z = sg * v + tt;
    z = (z >= 0.f) ? z : 0.2f * z;                 // leaky relu after max
    float f = feature[((size_t)b * NN + n) * CC + c];
    acc0 += mlpw[0 * 256 + c] * z + mlpw[0 * 256 + 128 + c] * f;
    acc1 += mlpw[1 * 256 + c] * z + mlpw[1 * 256 + 128 + c] * f;
    acc2 += mlpw[2 * 256 + c] * z + mlpw[2 * 256 + 128 + c] * f;
  }
  for (int m = 16; m >= 1; m >>= 1) {
    acc0 += __shfl_xor(acc0, m, 32);
    acc1 += __shfl_xor(acc1, m, 32);
    acc2 += __shfl_xor(acc2, m, 32);
  }
  if (lane == 0) {
    out[((size_t)b * 3 + 0) * NN + n] = acc0 + mlpb[0];
    out[((size_t)b * 3 + 1) * NN + n] = acc1 + mlpb[1];
    out[((size_t)b * 3 + 2) * NN + n] = acc2 + mlpb[2];
  }
}

__global__ void k_zero(float* __restrict__ p, int n) {
  int i = blockIdx.x * blockDim.x + threadIdx.x;
  if (i < n) p[i] = 0.f;
}

// =====================================================================
extern "C" void kernel_launch(void* const* d_in, const int* in_sizes, int n_in,
                              void* d_out, int out_size, void* d_ws, size_t ws_size,
                              hipStream_t stream) {
  (void)in_sizes; (void)n_in; (void)out_size; (void)ws_size;
  const float* pts   = (const float*)d_in[0];
  const float* feat  = (const float*)d_in[1];
  const float* inst  = (const float*)d_in[2];
  const float* w1    = (const float*)d_in[3];
  const float* w2    = (const float*)d_in[4];
  const float* convw = (const float*)d_in[5];
  const float* gamma = (const float*)d_in[6];
  const float* beta  = (const float*)d_in[7];
  const float* mlpw  = (const float*)d_in[8];
  const float* mlpb  = (const float*)d_in[9];
  float* out = (float*)d_out;

  char* ws = (char*)d_ws;
  size_t off = 0;
  auto take = [&](size_t bytes) -> char* {
    char* p = ws + off;
    off += (bytes + 255) & ~(size_t)255;
    return p;
  };
  int*   idx   = (int*)  take((size_t)KS * 4);
  float* gn    = (float*)take((size_t)BB * KS * CC * 4);
  float* kpb   = (float*)take((size_t)BB * KS * 3 * 4);
  float* fsb   = (float*)take((size_t)BB * KS * CC * 4);
  float* rnorm = (float*)take((size_t)BB * NN * 4);
  float* dist  = (float*)take((size_t)BB * NN * DPAD * 4);
  int*   tix   = (int*)  take((size_t)BB * NN * KN * 4);
  float* attn  = (float*)take((size_t)BB * NN * KN * 4);
  float* ymax  = (float*)take((size_t)BB * CC * NN * 4);
  float* ymin  = (float*)take((size_t)BB * CC * NN * 4);
  float* stats = (float*)take(16 * 4);
  float* gsum = stats, *gss = stats + 4, *gmu = stats + 8, *gscale = stats + 12;

  k_indices<<<1, 32, 0, stream>>>(idx);
  k_prep<<<(BB * KS) / 4, 128, 0, stream>>>(inst, pts, feat, idx, gn, kpb, fsb);
  k_rnorm<<<(BB * NN) / 4, 128, 0, stream>>>(inst, rnorm);
  dim3 gd(DPAD / 16, NN / 16, BB);
  k_dist_wmma<<<gd, 32, 0, stream>>>(inst, gn, rnorm, dist);
  k_topk_attn<<<(BB * NN) / 4, 128, 0, stream>>>(dist, w1, w2, tix, attn);
  k_zero<<<1, 32, 0, stream>>>(gsum, 8);   // gsum[4] + gss[4] contiguous
  k_conv_wmma<<<(BB * NN) / 4, 128, 0, stream>>>(pts, convw, tix, attn, kpb, fsb,
                                                 ymax, ymin, gsum, gss);
  k_stats<<<1, 32, 0, stream>>>(gsum, gss, gmu, gscale);
  k_final<<<(BB * NN) / 4, 128, 0, stream>>>(ymax, ymin, feat, gamma, beta,
                                             mlpw, mlpb, gmu, gscale, out);
}